// MMDLoss_69870527971660
// MI455X (gfx1250) — compile-verified
//
#include <hip/hip_runtime.h>
#include <hip/hip_bf16.h>

typedef float v2f __attribute__((ext_vector_type(2)));
typedef float v8f __attribute__((ext_vector_type(8)));

#define NTOT   4096
#define NHALF  2048
#define KDIM   1024
#define LDST   18            // LDS row stride in floats (padded, 8B-aligned)

__device__ __forceinline__ const float* rowptr(const float* __restrict__ s,
                                               const float* __restrict__ t, int i) {
    return (i < NHALF) ? (s + (size_t)i * KDIM) : (t + (size_t)(i - NHALF) * KDIM);
}

__device__ __forceinline__ float waveReduce(float v) {
    #pragma unroll
    for (int off = 16; off > 0; off >>= 1) v += __shfl_xor(v, off, 32);
    return v;
}

// ---------------------------------------------------------------- init
__global__ void mmd_init(float* ws) {
    ws[0] = 0.0f;   // sum(l2) accumulator
    ws[6] = 0.0f;   // signed kernel-sum accumulator
}

// ---------------------------------------------------------------- row ||x||^2
__global__ __launch_bounds__(256) void mmd_rowsq(const float* __restrict__ src,
                                                 const float* __restrict__ tgt,
                                                 float* __restrict__ sq) {
    const int row  = blockIdx.x;
    const int tid  = threadIdx.x;
    const int lane = tid & 31, wave = tid >> 5;
    const float4* p = (const float4*)rowptr(src, tgt, row);
    float4 v = p[tid];                       // 256 threads * 4 = 1024 elements
    float s = v.x * v.x + v.y * v.y + v.z * v.z + v.w * v.w;
    s = waveReduce(s);
    __shared__ float part[8];
    if (lane == 0) part[wave] = s;
    __syncthreads();
    if (tid == 0) {
        float t = 0.f;
        #pragma unroll
        for (int i = 0; i < 8; ++i) t += part[i];
        sq[row] = t;
    }
}

// ---------------------------------------------------------------- WMMA GEMM (both passes)
// Block: 256 threads = 8 waves -> 128x128 C tile. Wave grid 2x4, each wave 64x32.
__global__ __launch_bounds__(256) void mmd_gemm(const float* __restrict__ src,
                                                const float* __restrict__ tgt,
                                                const float* __restrict__ sq,
                                                float* __restrict__ ws,
                                                int pass2) {
    __shared__ float As[128 * LDST];
    __shared__ float Bs[128 * LDST];

    const int i0   = blockIdx.y * 128;
    const int j0   = blockIdx.x * 128;
    const int tid  = threadIdx.x;
    const int wave = tid >> 5;
    const int lane = tid & 31;
    const int lm   = lane & 15;
    const int hi   = lane >> 4;
    const int wr   = wave >> 2;          // 0..1
    const int wc   = wave & 3;           // 0..3
    const int rowbase = wr * 64;
    const int colbase = wc * 32;

    v8f acc[4][2];
    #pragma unroll
    for (int rg = 0; rg < 4; ++rg)
        #pragma unroll
        for (int cg = 0; cg < 2; ++cg) acc[rg][cg] = {};

    for (int k0 = 0; k0 < KDIM; k0 += 16) {
        __syncthreads();
        // stage A-panel rows i0..i0+127 and B-panel rows j0..j0+127, 16 k's each
        #pragma unroll
        for (int q = tid; q < 512; q += 256) {
            const int row = q >> 2;
            const int seg = (q & 3) << 2;
            const int o   = row * LDST + seg;
            {
                float4 v = *(const float4*)(rowptr(src, tgt, i0 + row) + k0 + seg);
                v2f lo = {v.x, v.y}, hiw = {v.z, v.w};
                *(v2f*)&As[o]     = lo;
                *(v2f*)&As[o + 2] = hiw;
            }
            {
                float4 v = *(const float4*)(rowptr(src, tgt, j0 + row) + k0 + seg);
                v2f lo = {v.x, v.y}, hiw = {v.z, v.w};
                *(v2f*)&Bs[o]     = lo;
                *(v2f*)&Bs[o + 2] = hiw;
            }
        }
        __syncthreads();

        #pragma unroll
        for (int kk = 0; kk < 16; kk += 4) {
            v2f a[4], b[2];
            #pragma unroll
            for (int rg = 0; rg < 4; ++rg)
                a[rg] = *(const v2f*)&As[(rowbase + rg * 16 + lm) * LDST + kk + 2 * hi];
            #pragma unroll
            for (int cg = 0; cg < 2; ++cg)
                b[cg] = *(const v2f*)&Bs[(colbase + cg * 16 + lm) * LDST + kk + 2 * hi];
            #pragma unroll
            for (int rg = 0; rg < 4; ++rg)
                #pragma unroll
                for (int cg = 0; cg < 2; ++cg)
                    acc[rg][cg] = __builtin_amdgcn_wmma_f32_16x16x4_f32(
                        false, a[rg], false, b[cg], (short)0, acc[rg][cg], false, false);
        }
    }

    // ---- epilogue: lane L, vgpr r of acc[rg][cg] holds C[i][j] with
    //      i = i0 + rowbase + rg*16 + 8*hi + r ,  j = j0 + colbase + cg*16 + lm
    float isq[4][8];
    #pragma unroll
    for (int rg = 0; rg < 4; ++rg)
        #pragma unroll
        for (int r = 0; r < 8; ++r)
            isq[rg][r] = sq[i0 + rowbase + rg * 16 + 8 * hi + r];
    float jsq[2];
    #pragma unroll
    for (int cg = 0; cg < 2; ++cg)
        jsq[cg] = sq[j0 + colbase + cg * 16 + lm];

    float part = 0.f;
    if (!pass2) {
        #pragma unroll
        for (int rg = 0; rg < 4; ++rg)
            #pragma unroll
            for (int cg = 0; cg < 2; ++cg)
                #pragma unroll
                for (int r = 0; r < 8; ++r) {
                    float l2 = fmaxf(isq[rg][r] + jsq[cg] - 2.0f * acc[rg][cg][r], 0.0f);
                    part += l2;
                }
        part = waveReduce(part);
        if (lane == 0) atomicAdd(&ws[0], part);
    } else {
        float ib[5];
        #pragma unroll
        for (int t = 0; t < 5; ++t) ib[t] = ws[1 + t];
        #pragma unroll
        for (int rg = 0; rg < 4; ++rg)
            #pragma unroll
            for (int cg = 0; cg < 2; ++cg)
                #pragma unroll
                for (int r = 0; r < 8; ++r) {
                    float l2 = fmaxf(isq[rg][r] + jsq[cg] - 2.0f * acc[rg][cg][r], 0.0f);
                    float kv = 0.f;
                    #pragma unroll
                    for (int t = 0; t < 5; ++t) kv += __expf(-l2 * ib[t]);
                    part += kv;
                }
        part = waveReduce(part);
        const float sign = ((i0 < NHALF) == (j0 < NHALF)) ? 1.0f : -1.0f;
        if (lane == 0) atomicAdd(&ws[6], sign * part);
    }
}

// ---------------------------------------------------------------- bandwidth
__global__ void mmd_bw(float* ws) {
    const float n  = (float)NTOT;
    float bandwidth = ws[0] / (n * n - n);
    bandwidth = bandwidth / 4.0f;             // kernel_mul^(kernel_num//2) = 2^2
    #pragma unroll
    for (int t = 0; t < 5; ++t)
        ws[1 + t] = 1.0f / (bandwidth * (float)(1 << t));
}

// ---------------------------------------------------------------- final scalar
__global__ void mmd_final(const float* ws, float* out) {
    out[0] = ws[6] / ((float)NHALF * (float)NHALF);
}

extern "C" void kernel_launch(void* const* d_in, const int* in_sizes, int n_in,
                              void* d_out, int out_size, void* d_ws, size_t ws_size,
                              hipStream_t stream) {
    const float* src = (const float*)d_in[0];
    const float* tgt = (const float*)d_in[1];
    float* out = (float*)d_out;
    float* ws  = (float*)d_ws;
    float* sq  = ws + 8;                       // 4096 floats of row norms

    mmd_init<<<1, 1, 0, stream>>>(ws);
    mmd_rowsq<<<NTOT, 256, 0, stream>>>(src, tgt, sq);

    dim3 grid(NTOT / 128, NTOT / 128);         // 32 x 32 tiles
    mmd_gemm<<<grid, 256, 0, stream>>>(src, tgt, sq, ws, 0);
    mmd_bw<<<1, 1, 0, stream>>>(ws);
    mmd_gemm<<<grid, 256, 0, stream>>>(src, tgt, sq, ws, 1);
    mmd_final<<<1, 1, 0, stream>>>(ws, out);
}